// two_step_47562467836029
// MI455X (gfx1250) — compile-verified
//
#include <hip/hip_runtime.h>
#include <hip/hip_bf16.h>

// ---------------------------------------------------------------------------
// Shallow-water timestep, gfx1250 (MI455X), wave32.
// 6 fused LDS-tiled stencil kernels; the Jacobi pressure sweeps use
// V_WMMA_F32_16X16X4_F32 (CDNA5 fp32 matrix op) for the column pass of the
// separable w1 stencil (ones3x3 - 9*delta).
// ---------------------------------------------------------------------------

#define NYI 2048
#define NXI 2048
#define NPTS (NYI * NXI)

#define DTc 0.8f
#define EPS_ETA 0.001f
#define MAN2 (0.055f * 0.055f)
#define DT_OVER_RHO (0.8f * 9.81f)     /* DT / RHO, RHO = 1/9.81 */
#define BETA_RHO (4.0f / 9.81f)        /* BETA * RHO */
#define INV3DX2 (1.0f / (3.0f * 5.0f * 5.0f))
#define INV12DX (1.0f / (12.0f * 5.0f))
#define INVDX (1.0f / 5.0f)
#define INVDX2 (1.0f / (5.0f * 5.0f))
#define DIAG (8.0f * INV3DX2)

typedef __attribute__((ext_vector_type(2))) float v2f;
typedef __attribute__((ext_vector_type(8))) float v8f;

__device__ __forceinline__ int clampi(int x, int lo, int hi) {
  return x < lo ? lo : (x > hi ? hi : x);
}
// _bc_u: zero left/right columns (x), copy top/bottom rows (y)
__device__ __forceinline__ float ld_bc_u(const float* f, int y, int x) {
  if (x < 0 || x >= NXI) return 0.0f;
  y = clampi(y, 0, NYI - 1);
  return f[y * NXI + x];
}
// _bc_v: copy left/right columns (x), zero top/bottom rows (y)
__device__ __forceinline__ float ld_bc_v(const float* f, int y, int x) {
  if (y < 0 || y >= NYI) return 0.0f;
  x = clampi(x, 0, NXI - 1);
  return f[y * NXI + x];
}
// _bc_eta: clamp (Neumann) on all sides
__device__ __forceinline__ float ld_bc_e(const float* f, int y, int x) {
  y = clampi(y, 0, NYI - 1);
  x = clampi(x, 0, NXI - 1);
  return f[y * NXI + x];
}

// w1 = (box3x3 - 9*center)/(3 dx^2)
__device__ __forceinline__ float conv_w1(const float* t, int i, int S) {
  float box = t[i - S - 1] + t[i - S] + t[i - S + 1]
            + t[i - 1]     + t[i]     + t[i + 1]
            + t[i + S - 1] + t[i + S] + t[i + S + 1];
  return (box - 9.0f * t[i]) * INV3DX2;
}
// w2: smoothed d/dx
__device__ __forceinline__ float conv_w2(const float* t, int i, int S) {
  return ((t[i - S + 1] - t[i - S - 1]) + 4.0f * (t[i + 1] - t[i - 1])
        + (t[i + S + 1] - t[i + S - 1])) * INV12DX;
}
// w3: smoothed d/dy
__device__ __forceinline__ float conv_w3(const float* t, int i, int S) {
  return ((t[i + S - 1] - t[i - S - 1]) + 4.0f * (t[i + S] - t[i - S])
        + (t[i + S + 1] - t[i - S + 1])) * INV12DX;
}
// wm: binomial smoother
__device__ __forceinline__ float conv_wm(const float* t, int i, int S) {
  return (t[i - S - 1] + t[i - S + 1] + t[i + S - 1] + t[i + S + 1]
        + 4.0f * (t[i - S] + t[i - 1] + t[i + 1] + t[i + S])
        + 16.0f * t[i]) * (1.0f / 36.0f);
}
// conv(a*b, w1) with a,b in different LDS tiles
__device__ __forceinline__ float conv_w1_prod(const float* a, int ia, int Sa,
                                              const float* b, int ib, int Sb) {
  float s = a[ia - Sa - 1] * b[ib - Sb - 1] + a[ia - Sa] * b[ib - Sb] + a[ia - Sa + 1] * b[ib - Sb + 1]
          + a[ia - 1] * b[ib - 1]           + a[ia] * b[ib]           + a[ia + 1] * b[ib + 1]
          + a[ia + Sa - 1] * b[ib + Sb - 1] + a[ia + Sa] * b[ib + Sb] + a[ia + Sa + 1] * b[ib + Sb + 1];
  return (s - 9.0f * a[ia] * b[ib]) * INV3DX2;
}
// Petrov-Galerkin coefficient at one point of padded field w
__device__ __forceinline__ float k_pg_at(const float* w, int i, int S, float amp) {
  float num = 0.25f * 5.0f * fabsf(amp * conv_w1(w, i, S));
  float den = EPS_ETA + 0.5f * (fabsf(conv_w2(w, i, S)) + fabsf(conv_w3(w, i, S))) * INVDX2;
  return num / den;
}

// ===========================================================================
// K1: predictor half step -> b_u_pre, b_v_pre (pre-friction, as padded later)
// ===========================================================================
__global__ __launch_bounds__(256) void k_predictor(
    const float* __restrict__ u, const float* __restrict__ v,
    const float* __restrict__ h, const float* __restrict__ k3,
    float* __restrict__ buPre, float* __restrict__ bvPre) {
  __shared__ float uu[36 * 36], vv[36 * 36], hpd[34 * 34];
  __shared__ float kuR[34 * 34], kvR[34 * 34], kuu[34 * 34], kvv[34 * 34];
  const int tid = threadIdx.x;
  const int by0 = blockIdx.y * 32, bx0 = blockIdx.x * 32;
  for (int i = tid; i < 36 * 36; i += 256) {
    int r = i / 36, c = i % 36, gy = by0 + r - 2, gx = bx0 + c - 2;
    uu[i] = ld_bc_u(u, gy, gx);
    vv[i] = ld_bc_v(v, gy, gx);
  }
  for (int i = tid; i < 34 * 34; i += 256) {
    int r = i / 34, c = i % 34;
    hpd[i] = ld_bc_e(h, by0 + r - 1, bx0 + c - 1);
  }
  __syncthreads();
  // k_u / k_v on tile + halo 1, with min(k3) and zero-pad (jnp.pad) outside
  for (int i = tid; i < 34 * 34; i += 256) {
    int r = i / 34, c = i % 34, gy = by0 + r - 1, gx = bx0 + c - 1;
    float aU = 0.f, aV = 0.f, mU = 0.f, mV = 0.f;
    if (gy >= 0 && gy < NYI && gx >= 0 && gx < NXI) {
      int wi = (r + 1) * 36 + (c + 1);
      float uc = uu[wi], vc = vv[wi];
      float amp = 0.5f * (fabsf(uc) + fabsf(vc)) * INVDX;
      aU = k_pg_at(uu, wi, 36, amp);
      aV = k_pg_at(vv, wi, 36, amp);
      float k3c = k3[gy * NXI + gx];
      mU = fminf(aU, k3c);
      mV = fminf(aV, k3c);
    }
    kuR[i] = aU; kvR[i] = aV; kuu[i] = mU; kvv[i] = mV;
  }
  __syncthreads();
  const int lx = tid & 31, ly0 = tid >> 5;
  for (int s = 0; s < 4; ++s) {
    int oy = ly0 + 8 * s, ox = lx;
    int gy = by0 + oy, gx = bx0 + ox, g = gy * NXI + gx;
    int wi = (oy + 2) * 36 + (ox + 2);
    int ki = (oy + 1) * 34 + (ox + 1);
    float uc = uu[wi], vc = vv[wi];
    float kx = 0.5f * (kuR[ki] * conv_w1(uu, wi, 36)
                     + conv_w1_prod(uu, wi, 36, kuu, ki, 34)
                     - uc * conv_w1(kuu, ki, 34));
    float ky = 0.5f * (kvR[ki] * conv_w1(vv, wi, 36)
                     + conv_w1_prod(vv, wi, 36, kvv, ki, 34)
                     - vc * conv_w1(kvv, ki, 34));
    float bu = 0.5f * DTc * (kx - uc * conv_w2(uu, wi, 36) - vc * conv_w3(uu, wi, 36))
             + uc - DTc * conv_w2(hpd, ki, 34);
    float bv = 0.5f * DTc * (ky - uc * conv_w2(vv, wi, 36) - vc * conv_w3(vv, wi, 36))
             + vc - DTc * conv_w3(hpd, ki, 34);
    buPre[g] = bu;
    bvPre[g] = bv;
  }
}

// ===========================================================================
// K2: corrector full step -> u_mid, v_mid (post-friction velocities)
// ===========================================================================
__global__ __launch_bounds__(256) void k_corrector(
    const float* __restrict__ u, const float* __restrict__ v,
    const float* __restrict__ h, const float* __restrict__ Hb,
    const float* __restrict__ k1, const float* __restrict__ k3,
    const float* __restrict__ buPre, const float* __restrict__ bvPre,
    float* __restrict__ uMid, float* __restrict__ vMid) {
  __shared__ float buu[36 * 36], bvv[36 * 36], hpd[34 * 34];
  __shared__ float pU[34 * 34], pV[34 * 34];
  __shared__ float kuR[34 * 34], kvR[34 * 34], kuu[34 * 34], kvv[34 * 34];
  const int tid = threadIdx.x;
  const int by0 = blockIdx.y * 32, bx0 = blockIdx.x * 32;
  for (int i = tid; i < 36 * 36; i += 256) {
    int r = i / 36, c = i % 36, gy = by0 + r - 2, gx = bx0 + c - 2;
    buu[i] = ld_bc_u(buPre, gy, gx);   // padded fields use PRE-friction b_u
    bvv[i] = ld_bc_v(bvPre, gy, gx);
  }
  for (int i = tid; i < 34 * 34; i += 256) {
    int r = i / 34, c = i % 34;
    hpd[i] = ld_bc_e(h, by0 + r - 1, bx0 + c - 1);
  }
  __syncthreads();
  for (int i = tid; i < 34 * 34; i += 256) {
    int r = i / 34, c = i % 34, gy = by0 + r - 1, gx = bx0 + c - 1;
    float aU = 0.f, aV = 0.f, mU = 0.f, mV = 0.f, fu = 0.f, fv = 0.f;
    if (gy >= 0 && gy < NYI && gx >= 0 && gx < NXI) {
      int wi = (r + 1) * 36 + (c + 1);
      int g = gy * NXI + gx;
      float preU = buu[wi], preV = bvv[wi];
      float d43 = powf(fmaxf(k1[g], Hb[g] + h[g]), 4.0f / 3.0f);
      float sg = sqrtf(preU * preU + preV * preV) * MAN2 / d43;
      float inv = 1.0f / (1.0f + sg * DT_OVER_RHO);
      fu = preU * inv;                 // post-friction b_u
      fv = preV * inv;
      float amp = 0.5f * (fabsf(fu) + fabsf(fv)) * INVDX;
      aU = k_pg_at(buu, wi, 36, amp);
      aV = k_pg_at(bvv, wi, 36, amp);
      float k3c = k3[g];
      mU = fminf(aU, k3c);
      mV = fminf(aV, k3c);
    }
    pU[i] = fu; pV[i] = fv;
    kuR[i] = aU; kvR[i] = aV; kuu[i] = mU; kvv[i] = mV;
  }
  __syncthreads();
  const int lx = tid & 31, ly0 = tid >> 5;
  for (int s = 0; s < 4; ++s) {
    int oy = ly0 + 8 * s, ox = lx;
    int gy = by0 + oy, gx = bx0 + ox, g = gy * NXI + gx;
    int wi = (oy + 2) * 36 + (ox + 2);
    int ki = (oy + 1) * 34 + (ox + 1);
    float fu = pU[ki], fv = pV[ki];
    float kx = 0.5f * (kuR[ki] * conv_w1(buu, wi, 36)
                     + conv_w1_prod(buu, wi, 36, kuu, ki, 34)
                     - fu * conv_w1(kuu, ki, 34));
    float ky = 0.5f * (kvR[ki] * conv_w1(bvv, wi, 36)
                     + conv_w1_prod(bvv, wi, 36, kvv, ki, 34)
                     - fv * conv_w1(kvv, ki, 34));
    float un = u[g] + DTc * (kx - fu * conv_w2(buu, wi, 36) - fv * conv_w3(buu, wi, 36))
             - DTc * conv_w2(hpd, ki, 34);
    float vn = v[g] + DTc * (ky - fu * conv_w2(bvv, wi, 36) - fv * conv_w3(bvv, wi, 36))
             - DTc * conv_w3(hpd, ki, 34);
    float d43 = powf(fmaxf(k1[g], Hb[g] + h[g]), 4.0f / 3.0f);
    float sg = sqrtf(un * un + vn * vn) * MAN2 / d43;
    float inv = 1.0f / (1.0f + sg * DT_OVER_RHO);
    uMid[g] = un * inv;
    vMid[g] = vn * inv;
  }
}

// ===========================================================================
// K3: wave-equation RHS  b
// ===========================================================================
__global__ __launch_bounds__(256) void k_rhs(
    const float* __restrict__ uMid, const float* __restrict__ vMid,
    const float* __restrict__ h, const float* __restrict__ Hb,
    const float* __restrict__ k1, const float* __restrict__ k2,
    const float* __restrict__ k3, const float* __restrict__ src,
    const float* __restrict__ difh, float* __restrict__ bOut) {
  __shared__ float ep[36 * 36];
  __shared__ float un[34 * 34], vn[34 * 34], dp[34 * 34];
  __shared__ float ku[34 * 34], kR[34 * 34];
  const int tid = threadIdx.x;
  const int by0 = blockIdx.y * 32, bx0 = blockIdx.x * 32;
  for (int i = tid; i < 36 * 36; i += 256) {
    int r = i / 36, c = i % 36;
    int cy = clampi(by0 + r - 2, 0, NYI - 1), cx = clampi(bx0 + c - 2, 0, NXI - 1);
    int g = cy * NXI + cx;
    ep[i] = fmaxf(k2[g], Hb[g] + h[g]);   // bc_eta(eta1) via clamp
  }
  for (int i = tid; i < 34 * 34; i += 256) {
    int r = i / 34, c = i % 34, gy = by0 + r - 1, gx = bx0 + c - 1;
    un[i] = ld_bc_u(uMid, gy, gx);
    vn[i] = ld_bc_v(vMid, gy, gx);
    dp[i] = ld_bc_e(difh, gy, gx);
  }
  __syncthreads();
  for (int i = tid; i < 34 * 34; i += 256) {
    int r = i / 34, c = i % 34, gy = by0 + r - 1, gx = bx0 + c - 1;
    float kr = 0.f, km = 0.f;
    if (gy >= 0 && gy < NYI && gx >= 0 && gx < NXI) {
      int wi = (r + 1) * 36 + (c + 1);
      float amp = 0.5f * (fabsf(un[i]) + fabsf(vn[i])) * INVDX;
      kr = k_pg_at(ep, wi, 36, amp);
      km = fminf(kr, k3[gy * NXI + gx]);
    }
    kR[i] = kr;
    ku[i] = km;
  }
  __syncthreads();
  const int lx = tid & 31, ly0 = tid >> 5;
  for (int s = 0; s < 4; ++s) {
    int oy = ly0 + 8 * s, ox = lx;
    int gy = by0 + oy, gx = bx0 + ox, g = gy * NXI + gx;
    int wi = (oy + 2) * 36 + (ox + 2);
    int ki = (oy + 1) * 34 + (ox + 1);
    float uc = un[ki], vc = vn[ki];
    float e1 = ep[wi];
    float e2 = fmaxf(k1[g], Hb[g] + h[g]);
    float pg = 0.5f * (kR[ki] * conv_w1(ep, wi, 36)
                     + conv_w1_prod(ep, wi, 36, ku, ki, 34)
                     - e1 * conv_w1(ku, ki, 34));
    float bval = BETA_RHO * (-conv_w2(ep, wi, 36) * uc - conv_w3(ep, wi, 36) * vc
                             - e1 * (conv_w2(un, ki, 34) + conv_w3(vn, ki, 34))
                             + pg - conv_wm(dp, ki, 34) * (1.0f / DTc) + src[g])
               / (DTc * e2);
    bOut[g] = bval;
  }
}

// ===========================================================================
// K4: one Jacobi sweep.  w1-stencil via V_WMMA_F32_16X16X4_F32:
//   rowsum (LDS, VALU)  then  columns: D(16x16) = A_band(16x20) * RS(20x16)
//   as 5 chained K=4 WMMA accumulations (banded 0/1 A generated per lane).
// rs is padded to 36 rows (rows 34..35 zeroed) so every B-operand ds_load is
// unconditional and in-bounds: no exec-mask divergence inside the WMMA chain
// (A is 0 there anyway, and 0*finite keeps the accumulator exact).
// Block = 256 threads = 8 waves; each wave owns one 16x16 tile of a 32x64
// block region.  A layout: VGPR0 = K0|K2 (lane<16|>=16), VGPR1 = K1|K3.
// D layout: VGPR j = rows j | j+8.
// ===========================================================================
__global__ __launch_bounds__(256) void k_jacobi(
    const float* __restrict__ hhIn, const float* __restrict__ bsrc,
    const float* __restrict__ Hb, const float* __restrict__ h,
    const float* __restrict__ k1, float* __restrict__ hhOut) {
  __shared__ float hpd[34 * 66];   // bc_eta(hh): rows -1..32, cols -1..64
  __shared__ float rs[36 * 64];    // horizontal 3-sums, rows 34..35 zero pad
  const int tid = threadIdx.x;
  const int by0 = blockIdx.y * 32, bx0 = blockIdx.x * 64;
  for (int i = tid; i < 34 * 66; i += 256) {
    int r = i / 66, c = i % 66;
    int gy = clampi(by0 + r - 1, 0, NYI - 1), gx = clampi(bx0 + c - 1, 0, NXI - 1);
    hpd[i] = hhIn[gy * NXI + gx];
  }
  __syncthreads();
  for (int i = tid; i < 34 * 64; i += 256) {
    int r = i / 64, c = i % 64;
    int base = r * 66 + c;
    rs[i] = hpd[base] + hpd[base + 1] + hpd[base + 2];
  }
  if (tid < 128) rs[34 * 64 + tid] = 0.0f;   // zero pad rows 34..35
  __syncthreads();

  const int lane = tid & 31;
  const int wv = tid >> 5;          // 0..7 waves
  const int ty = wv >> 2;           // tile row 0..1
  const int tx = wv & 3;            // tile col 0..3
  const int n = lane & 15;          // column within tile / A-row M
  const int hi = lane >> 4;         // 0: K0/K1 half, 1: K2/K3 half
  const int rowbase = ty * 16;
  const int colbase = tx * 16 + n;

  v8f acc = {0.f, 0.f, 0.f, 0.f, 0.f, 0.f, 0.f, 0.f};
#pragma unroll
  for (int c4 = 0; c4 < 5; ++c4) {
    int kx = 4 * c4 + 2 * hi;       // K of .x component this lane holds
    int ky = kx + 1;                // K of .y component
    v2f av, bv;
    // banded A: A[m][K] = 1 iff K in {m, m+1, m+2}   (m = n here)
    av.x = (kx >= n && kx <= n + 2) ? 1.0f : 0.0f;
    av.y = (ky >= n && ky <= n + 2) ? 1.0f : 0.0f;
    // B[K][col] = rs row (rowbase + K); rows >= 34 are zero pad
    bv.x = rs[(rowbase + kx) * 64 + colbase];
    bv.y = rs[(rowbase + ky) * 64 + colbase];
    acc = __builtin_amdgcn_wmma_f32_16x16x4_f32(
        false, av, false, bv, (short)0, acc, false, false);
  }

#pragma unroll
  for (int j = 0; j < 8; ++j) {
    int oy = rowbase + j + 8 * hi;      // block-local row (D layout)
    int ox = tx * 16 + n;
    int gy = by0 + oy, gx = bx0 + ox, g = gy * NXI + gx;
    float center = hpd[(oy + 1) * 66 + (ox + 1)];
    float conv = (acc[j] - 9.0f * center) * INV3DX2;   // conv(hh, w1)
    float e2 = fmaxf(k1[g], Hb[g] + h[g]);
    float coef = BETA_RHO / (DTc * DTc * e2);
    float denom = DIAG + coef;
    hhOut[g] = center - (-conv + coef * center) / denom + bsrc[g] / denom;
  }
}

// ===========================================================================
// K5: finalize outputs
// ===========================================================================
__global__ __launch_bounds__(256) void k_final(
    const float* __restrict__ uMid, const float* __restrict__ vMid,
    const float* __restrict__ h, const float* __restrict__ hh,
    float* __restrict__ oU, float* __restrict__ oV,
    float* __restrict__ oH, float* __restrict__ oDif) {
  __shared__ float hpd[34 * 34];
  const int tid = threadIdx.x;
  const int by0 = blockIdx.y * 32, bx0 = blockIdx.x * 32;
  for (int i = tid; i < 34 * 34; i += 256) {
    int r = i / 34, c = i % 34;
    hpd[i] = ld_bc_e(hh, by0 + r - 1, bx0 + c - 1);
  }
  __syncthreads();
  const int lx = tid & 31, ly0 = tid >> 5;
  for (int s = 0; s < 4; ++s) {
    int oy = ly0 + 8 * s, ox = lx;
    int gy = by0 + oy, gx = bx0 + ox, g = gy * NXI + gx;
    int ki = (oy + 1) * 34 + (ox + 1);
    float c = hpd[ki];
    oU[g] = uMid[g] - conv_w2(hpd, ki, 34) * DT_OVER_RHO;
    oV[g] = vMid[g] - conv_w3(hpd, ki, 34) * DT_OVER_RHO;
    oH[g] = h[g] + c;
    oDif[g] = c;                      // dif_h == hh exactly
  }
}

// ===========================================================================
extern "C" void kernel_launch(void* const* d_in, const int* in_sizes, int n_in,
                              void* d_out, int out_size, void* d_ws, size_t ws_size,
                              hipStream_t stream) {
  (void)in_sizes; (void)n_in; (void)out_size; (void)ws_size;
  const float* u    = (const float*)d_in[0];
  const float* v    = (const float*)d_in[1];
  const float* Hb   = (const float*)d_in[2];
  const float* h    = (const float*)d_in[3];
  const float* k1   = (const float*)d_in[4];
  const float* k2   = (const float*)d_in[5];
  const float* k3   = (const float*)d_in[6];
  const float* src  = (const float*)d_in[7];
  const float* difh = (const float*)d_in[8];
  const float* hh0  = (const float*)d_in[9];
  // weights d_in[10..13] are the fixed stencils; baked into the kernels.

  float* out  = (float*)d_out;
  float* oU   = out;
  float* oV   = out + (size_t)NPTS;
  float* oH   = out + 2 * (size_t)NPTS;
  float* oHH  = out + 3 * (size_t)NPTS;
  float* oB   = out + 4 * (size_t)NPTS;
  float* oDif = out + 5 * (size_t)NPTS;

  float* ws    = (float*)d_ws;       // needs 5 * 16 MiB
  float* buPre = ws;
  float* bvPre = ws + (size_t)NPTS;
  float* uMid  = ws + 2 * (size_t)NPTS;
  float* vMid  = ws + 3 * (size_t)NPTS;
  float* hhT   = ws + 4 * (size_t)NPTS;

  dim3 blk(256);
  dim3 g32(NXI / 32, NYI / 32);
  dim3 gJ(NXI / 64, NYI / 32);

  k_predictor<<<g32, blk, 0, stream>>>(u, v, h, k3, buPre, bvPre);
  k_corrector<<<g32, blk, 0, stream>>>(u, v, h, Hb, k1, k3, buPre, bvPre, uMid, vMid);
  k_rhs<<<g32, blk, 0, stream>>>(uMid, vMid, h, Hb, k1, k2, k3, src, difh, oB);
  k_jacobi<<<gJ, blk, 0, stream>>>(hh0, oB, Hb, h, k1, hhT);
  k_jacobi<<<gJ, blk, 0, stream>>>(hhT, oB, Hb, h, k1, oHH);
  k_final<<<g32, blk, 0, stream>>>(uMid, vMid, h, oHH, oU, oV, oH, oDif);
}